// SO8TRotationGate_25838523252900
// MI455X (gfx1250) — compile-verified
//
#include <hip/hip_runtime.h>
#include <hip/hip_bf16.h>

typedef float v2f __attribute__((ext_vector_type(2)));
typedef float v4f __attribute__((ext_vector_type(4)));
typedef float v8f __attribute__((ext_vector_type(8)));

#define D_DIM 1024
#define NBLK  128     // number of 8x8 blocks
#define NTILE 64      // number of 16x16 block-diagonal tiles (pairs of blocks)
#define ROWS  32768   // B*T

// ---------------------------------------------------------------------------
// Kernel 1: compute R[n] = expm(theta[n] - theta[n]^T) (truncated Taylor,
// matching the reference), then pack pairs of R blocks into the WMMA
// B-operand layout for v_wmma_f32_16x16x4_f32.
//
// Packed layout in ws: wsB[t][lane][j], j = k*2 + p, where for K-chunk k the
// B operand VGPR p on lane L holds M[4k + p + 2*(L>>4)][L & 15] of the 16x16
// block-diagonal matrix M = diag(R[2t], R[2t+1]).  Each lane's 8 floats are
// contiguous -> two b128 loads per tile in the apply kernel.
// ---------------------------------------------------------------------------
__global__ __launch_bounds__(128) void so8_prepare(const float* __restrict__ theta,
                                                   float* __restrict__ wsB) {
  __shared__ float sR[NBLK][64];
  __shared__ float snorm[NBLK];

  const int n = threadIdx.x;  // one 8x8 block per thread
  float A[64], R[64], Ap[64], T[64];

  // A = theta - theta^T, then += 1e-6 * I
  #pragma unroll
  for (int e = 0; e < 8; ++e)
    #pragma unroll
    for (int o = 0; o < 8; ++o)
      A[e * 8 + o] = theta[n * 64 + e * 8 + o] - theta[n * 64 + o * 8 + e];
  #pragma unroll
  for (int i = 0; i < 8; ++i) A[i * 9] += 1e-6f;

  // Frobenius norm, clamp-scale
  float s2 = 0.f;
  #pragma unroll
  for (int i = 0; i < 64; ++i) s2 += A[i] * A[i];
  const float nrm = sqrtf(s2);
  snorm[n] = nrm;
  const float scale = fminf(nrm, 1.0f);
  const float inv = 1.0f / (scale + 1e-8f);
  #pragma unroll
  for (int i = 0; i < 64; ++i) A[i] *= inv;

  // Taylor: R = I + sum_{i=1..10} A^i / i!
  #pragma unroll
  for (int i = 0; i < 64; ++i) { R[i] = (i % 9 == 0) ? 1.0f : 0.0f; Ap[i] = A[i]; }
  const float rfact[10] = {1.f, 0.5f, 1.f/6.f, 1.f/24.f, 1.f/120.f, 1.f/720.f,
                           1.f/5040.f, 1.f/40320.f, 1.f/362880.f, 1.f/3628800.f};
  for (int it = 0; it < 10; ++it) {
    #pragma unroll
    for (int i = 0; i < 64; ++i) R[i] += Ap[i] * rfact[it];
    if (it < 9) {
      for (int r = 0; r < 8; ++r)
        for (int c = 0; c < 8; ++c) {
          float acc = 0.f;
          #pragma unroll
          for (int k = 0; k < 8; ++k) acc += Ap[r * 8 + k] * A[k * 8 + c];
          T[r * 8 + c] = acc;
        }
      #pragma unroll
      for (int i = 0; i < 64; ++i) Ap[i] = T[i];
    }
  }

  __syncthreads();
  // global decision: square R iff max over ALL blocks of min(norm,1) >= 1
  bool do_sq = false;
  for (int i = 0; i < NBLK; ++i) do_sq = do_sq || (snorm[i] >= 1.0f);
  if (do_sq) {
    for (int r = 0; r < 8; ++r)
      for (int c = 0; c < 8; ++c) {
        float acc = 0.f;
        #pragma unroll
        for (int k = 0; k < 8; ++k) acc += R[r * 8 + k] * R[k * 8 + c];
        T[r * 8 + c] = acc;
      }
    #pragma unroll
    for (int i = 0; i < 64; ++i) R[i] = T[i];
  }

  #pragma unroll
  for (int i = 0; i < 64; ++i) sR[n][i] = R[i];
  __syncthreads();

  // Pack B-operand layout: 64 tiles * 32 lanes * 8 floats = 16384 floats
  for (int idx = threadIdx.x; idx < NTILE * 32 * 8; idx += 128) {
    const int j    = idx & 7;           // j = k*2 + p
    const int lane = (idx >> 3) & 31;
    const int t    = idx >> 8;
    const int k = j >> 1, p = j & 1;
    const int row = 4 * k + p + 2 * (lane >> 4);  // K index within 16x16 M
    const int col = lane & 15;                    // N index
    float v = 0.f;
    if ((row >> 3) == (col >> 3))                 // block-diagonal
      v = sR[2 * t + (row >> 3)][(row & 7) * 8 + (col & 7)];
    wsB[idx] = v;  // idx == t*256 + lane*8 + j
  }
}

// ---------------------------------------------------------------------------
// Kernel 2: streaming y = x * blockdiag(R)  via v_wmma_f32_16x16x4_f32.
// Each wave: 16 rows x 64 cols (4 tiles of 16x16, 4 chained WMMAs per tile).
// ---------------------------------------------------------------------------
__global__ __launch_bounds__(256) void so8_apply(const float* __restrict__ x,
                                                 const float* __restrict__ wsB,
                                                 float* __restrict__ y) {
  const int lane    = threadIdx.x & 31;
  const int wave    = threadIdx.x >> 5;
  const int rowBase = (blockIdx.x >> 1) * 16;
  const int colBase = (blockIdx.x & 1) * 512 + wave * 64;
  const int mrow    = lane & 15;  // A-matrix row / D-matrix column
  const int hsel    = lane >> 4;  // lane-half select

  const float* __restrict__ xrow = x + (size_t)(rowBase + mrow) * D_DIM;

  #pragma unroll
  for (int ct = 0; ct < 4; ++ct) {
    const int c = colBase + ct * 16;
    const int t = c >> 4;

    // B operand: 8 contiguous floats per lane (hot in L0/L2, 64KB total)
    const float* bp = wsB + (size_t)t * 256 + (size_t)lane * 8;
    const v4f b0 = *(const v4f*)(bp);
    const v4f b1 = *(const v4f*)(bp + 4);
    const v2f B0 = {b0.x, b0.y};
    const v2f B1 = {b0.z, b0.w};
    const v2f B2 = {b1.x, b1.y};
    const v2f B3 = {b1.z, b1.w};

    // A operand: per K-chunk, lane L holds K = 4k + 2*hsel (+1) of row mrow
    const float* ap = xrow + c + 2 * hsel;
    const v2f a0 = *(const v2f*)(ap + 0);
    const v2f a1 = *(const v2f*)(ap + 4);
    const v2f a2 = *(const v2f*)(ap + 8);
    const v2f a3 = *(const v2f*)(ap + 12);

    v8f acc = {};
    acc = __builtin_amdgcn_wmma_f32_16x16x4_f32(false, a0, false, B0, (short)0, acc, false, false);
    acc = __builtin_amdgcn_wmma_f32_16x16x4_f32(false, a1, false, B1, (short)0, acc, false, false);
    acc = __builtin_amdgcn_wmma_f32_16x16x4_f32(false, a2, false, B2, (short)0, acc, false, false);
    acc = __builtin_amdgcn_wmma_f32_16x16x4_f32(false, a3, false, B3, (short)0, acc, false, false);

    // D layout: VGPR v -> row rowBase + v + 8*hsel, col c + mrow
    float* yp = y + (size_t)(rowBase + 8 * hsel) * D_DIM + c + mrow;
    #pragma unroll
    for (int v = 0; v < 8; ++v)
      yp[(size_t)v * D_DIM] = acc[v];
  }
}

extern "C" void kernel_launch(void* const* d_in, const int* in_sizes, int n_in,
                              void* d_out, int out_size, void* d_ws, size_t ws_size,
                              hipStream_t stream) {
  const float* x     = (const float*)d_in[0];
  const float* theta = (const float*)d_in[1];
  float* yv  = (float*)d_out;
  float* wsB = (float*)d_ws;   // needs 64KB: 64 tiles * 32 lanes * 8 floats

  so8_prepare<<<1, 128, 0, stream>>>(theta, wsB);
  // 2048 row-tiles * 2 half-row blocks; 256 threads = 8 waves * (16 rows x 64 cols)
  so8_apply<<<4096, 256, 0, stream>>>(x, wsB, yv);
}